// PersistenceImageLayer_37907381354923
// MI455X (gfx1250) — compile-verified
//
#include <hip/hip_runtime.h>

typedef float v8f __attribute__((ext_vector_type(8)));
typedef float v2f __attribute__((ext_vector_type(2)));

namespace {
constexpr int   kRes    = 20;
constexpr int   kP      = 2048;
constexpr int   kBD     = 96;          // 32 batches * 3 dims
constexpr int   kWaves  = 8;
constexpr int   kKSlice = kP / kWaves; // 256, multiple of 4
constexpr float kSigma  = 0.1f;
}

// image[bd] = A^T * B where A[p,x] = w_p * exp(-(x-b_p)^2/2s^2),
//             B[p,y] = exp(-(y-pers_p)^2/2s^2), done with V_WMMA_F32_16X16X4_F32.
// A-fragment lane layout (16x4 f32): lane L holds rows M=L&15; VGPR0/1 hold
// K = (L>>4)*2 and (L>>4)*2+1. B-fragment (4x16) mirrors this with N=L&15.
__global__ __launch_bounds__(256)
void persistence_image_wmma(const float* __restrict__ pd, float* __restrict__ out)
{
    const int lane = threadIdx.x & 31;
    const int wave = threadIdx.x >> 5;   // 0..7, each owns a K-slice of 256 pts
    const int bd   = blockIdx.x;         // 0..95

    const float inv2s2 = 1.0f / (2.0f * kSigma * kSigma);
    const float step   = 1.0f / (float)(kRes - 1);   // linspace(0,1,20) spacing

    const float2* __restrict__ base = (const float2*)(pd + (size_t)bd * kP * 2);

    const int half = lane >> 4;   // selects which K-pair of the chunk this lane feeds
    const int mn   = lane & 15;   // M for A-fragments, N for B-fragments

    // Pixel coordinates for row-tile 0/1 (same values serve y, since Rx==Ry).
    const float c0 = (float)mn * step;          // pixels 0..15
    const float c1 = (float)(mn + 16) * step;   // pixels 16..31 (only 16..19 kept)

    v8f acc00 = {}; v8f acc01 = {}; v8f acc10 = {}; v8f acc11 = {};

    const int kBeg = wave * kKSlice;
    const int kEnd = kBeg + kKSlice;

    for (int k = kBeg; k < kEnd; k += 4) {
        const int p0 = k + half * 2;          // this lane supplies K-pair (p0, p0+1)
        const float2 q0 = base[p0];
        const float2 q1 = base[p0 + 1];

        const float b0 = q0.x, pr0 = q0.y - q0.x;   // birth, persistence
        const float b1 = q1.x, pr1 = q1.y - q1.x;
        const float w0 = fmaxf(pr0, 0.0f);          // eff_w = pers if pers>0 else 0
        const float w1 = fmaxf(pr1, 0.0f);

        float d;
        d = c0 - b0;  const float gx00 = __expf(-d * d * inv2s2);
        d = c0 - b1;  const float gx01 = __expf(-d * d * inv2s2);
        d = c1 - b0;  const float gx10 = __expf(-d * d * inv2s2);
        d = c1 - b1;  const float gx11 = __expf(-d * d * inv2s2);
        d = c0 - pr0; const float gy00 = __expf(-d * d * inv2s2);
        d = c0 - pr1; const float gy01 = __expf(-d * d * inv2s2);
        d = c1 - pr0; const float gy10 = __expf(-d * d * inv2s2);
        d = c1 - pr1; const float gy11 = __expf(-d * d * inv2s2);

        v2f A0; A0.x = w0 * gx00; A0.y = w1 * gx01;   // row-tile 0, weighted
        v2f A1; A1.x = w0 * gx10; A1.y = w1 * gx11;   // row-tile 1, weighted
        v2f B0; B0.x = gy00;      B0.y = gy01;        // col-tile 0
        v2f B1; B1.x = gy10;      B1.y = gy11;        // col-tile 1

        // D = A x B + C, full fp32, K=4 per issue
        acc00 = __builtin_amdgcn_wmma_f32_16x16x4_f32(false, A0, false, B0, (short)0, acc00, false, false);
        acc01 = __builtin_amdgcn_wmma_f32_16x16x4_f32(false, A0, false, B1, (short)0, acc01, false, false);
        acc10 = __builtin_amdgcn_wmma_f32_16x16x4_f32(false, A1, false, B0, (short)0, acc10, false, false);
        acc11 = __builtin_amdgcn_wmma_f32_16x16x4_f32(false, A1, false, B1, (short)0, acc11, false, false);
    }

    // Deterministic cross-wave reduction through LDS (no float atomics).
    __shared__ float partial[kWaves][4][8][32];   // 32 KB of the 320 KB WGP pool
    #pragma unroll
    for (int r = 0; r < 8; ++r) {
        partial[wave][0][r][lane] = acc00[r];
        partial[wave][1][r][lane] = acc01[r];
        partial[wave][2][r][lane] = acc10[r];
        partial[wave][3][r][lane] = acc11[r];
    }
    __syncthreads();

    // Waves 0..3 each finalize one 16x16 tile; C/D layout: lane L holds N=L&15,
    // VGPR r holds M = r + (L>>4)*8.
    if (wave < 4) {
        const int tm = wave >> 1;
        const int tn = wave & 1;
        #pragma unroll
        for (int r = 0; r < 8; ++r) {
            float s = 0.0f;
            #pragma unroll
            for (int w = 0; w < kWaves; ++w) s += partial[w][wave][r][lane];
            const int x = tm * 16 + half * 8 + r;
            const int y = tn * 16 + mn;
            if (x < kRes && y < kRes)
                out[(size_t)bd * (kRes * kRes) + x * kRes + y] = s;
        }
    }
}

extern "C" void kernel_launch(void* const* d_in, const int* in_sizes, int n_in,
                              void* d_out, int out_size, void* d_ws, size_t ws_size,
                              hipStream_t stream)
{
    (void)in_sizes; (void)n_in; (void)out_size; (void)d_ws; (void)ws_size;
    const float* pd = (const float*)d_in[0];
    float* out = (float*)d_out;
    persistence_image_wmma<<<kBD, 256, 0, stream>>>(pd, out);
}